// GINLayer_74491912781908
// MI455X (gfx1250) — compile-verified
//
#include <hip/hip_runtime.h>
#include <hip/hip_bf16.h>

// ---------------------------------------------------------------------------
// GIN layer for MI455X (gfx1250, wave32):
//   agg = (1+eps)*x + segment_sum(x[col], row)
//   h1  = agg @ W1^T + b1            (fp32 WMMA 16x16x4, BN stats fused)
//   h   = relu((h1-mu)*rsqrt(var+e)*gamma + beta)   (folded to scale/shift)
//   out = h @ W2^T + b2              (fp32 WMMA, BN+ReLU fused into A-stage)
// ---------------------------------------------------------------------------

#define N_NODES 100000
#define N_EDGES 1600000
#define F_IN    128
#define F_OUT   128
#define BN_EPS  1e-5f

typedef __attribute__((ext_vector_type(2))) float v2f;
typedef __attribute__((ext_vector_type(8))) float v8f;

// LDS row strides (floats), padded so WMMA operand reads hit 32 distinct banks
#define SA 132
#define SW 132

// ---------------------------------------------------------------------------
// Kernel 1: agg[i] = (1+eps)*x[i]; block 0 also zeroes the BN stat accums.
// ---------------------------------------------------------------------------
__global__ __launch_bounds__(256)
void gin_init_kernel(const float* __restrict__ x, const float* __restrict__ epsp,
                     float* __restrict__ agg, float* __restrict__ stats)
{
    if (blockIdx.x == 0 && threadIdx.x < 2 * F_OUT)
        stats[threadIdx.x] = 0.0f;            // sum[128] + sumsq[128]

    const float s = 1.0f + *epsp;
    size_t t = (size_t)blockIdx.x * blockDim.x + threadIdx.x;   // float4 index
    const size_t total4 = (size_t)N_NODES * F_IN / 4;
    if (t < total4) {
        float4 v = ((const float4*)x)[t];
        v.x *= s; v.y *= s; v.z *= s; v.w *= s;
        ((float4*)agg)[t] = v;
    }
}

// ---------------------------------------------------------------------------
// Kernel 2: edge scatter. One wave32 per edge; lane l handles features
// [4l, 4l+4). 819 MB of gathers (L2-resident: x is 51 MB < 192 MB L2) plus
// global_atomic_add_f32 scatter.
// ---------------------------------------------------------------------------
__global__ __launch_bounds__(256)
void gin_scatter_kernel(const float* __restrict__ x,
                        const long long* __restrict__ row,
                        const long long* __restrict__ col,
                        float* __restrict__ agg)
{
    size_t t = (size_t)blockIdx.x * blockDim.x + threadIdx.x;
    const int e    = (int)(t >> 5);      // whole wave shares one edge
    const int lane = (int)(t & 31);
    if (e >= N_EDGES) return;

    const int r = (int)row[e];
    const int c = (int)col[e];
    const float4 v = ((const float4*)(x + (size_t)c * F_IN))[lane];
    float* dst = agg + (size_t)r * F_IN + lane * 4;
    atomicAdd(dst + 0, v.x);
    atomicAdd(dst + 1, v.y);
    atomicAdd(dst + 2, v.z);
    atomicAdd(dst + 3, v.w);
}

// ---------------------------------------------------------------------------
// GEMM kernel: C[16 x 128] tile per block, 8 waves, one 16x16 col-tile per
// wave, K=128 via 32x v_wmma_f32_16x16x4_f32. FUSE=1 applies per-channel
// scale/shift + ReLU to A while staging (BN for layer 2). FUSE=0 accumulates
// per-channel sum/sumsq of the result (BN stats for layer 1).
// ---------------------------------------------------------------------------
template<int FUSE>
__global__ __launch_bounds__(256)
void gin_gemm_kernel(const float* __restrict__ Ain,    // [N,128]
                     const float* __restrict__ W,      // [128,128] (out,in)
                     const float* __restrict__ bias,   // [128]
                     const float* __restrict__ scale,  // [128] (FUSE only)
                     const float* __restrict__ shift,  // [128] (FUSE only)
                     float* __restrict__ Cout,         // [N,128]
                     float* __restrict__ sumOut,       // [128] (!FUSE)
                     float* __restrict__ sumsqOut)     // [128] (!FUSE)
{
    __shared__ float lA[16 * SA];      // A panel, 16 rows x 128 K
    __shared__ float lW[32 * SW];      // W chunk, transposed: [k][n]

    const int lane    = threadIdx.x & 31;
    const int wave    = threadIdx.x >> 5;       // 0..7 -> output col tile
    const int rowBase = blockIdx.x * 16;        // 6250 * 16 == 100000 exactly

    // ---- stage A panel (optionally BN+ReLU transformed) -------------------
    {
        const int m0 = threadIdx.x >> 5;        // rows 0..7, then +8
        const int k4 = (threadIdx.x & 31) * 4;
        #pragma unroll
        for (int i = 0; i < 2; ++i) {
            const int m = m0 + 8 * i;
            float4 v = *(const float4*)(Ain + (size_t)(rowBase + m) * F_IN + k4);
            if (FUSE) {
                v.x = fmaxf(v.x * scale[k4 + 0] + shift[k4 + 0], 0.0f);
                v.y = fmaxf(v.y * scale[k4 + 1] + shift[k4 + 1], 0.0f);
                v.z = fmaxf(v.z * scale[k4 + 2] + shift[k4 + 2], 0.0f);
                v.w = fmaxf(v.w * scale[k4 + 3] + shift[k4 + 3], 0.0f);
            }
            lA[m * SA + k4 + 0] = v.x;
            lA[m * SA + k4 + 1] = v.y;
            lA[m * SA + k4 + 2] = v.z;
            lA[m * SA + k4 + 3] = v.w;
        }
    }

    v8f c = {};                                   // f32 accumulator tile
    const int nCol  = wave * 16 + (lane & 15);    // this lane's output column
    const int kHalf = (lane >> 4) * 2;            // K sub-offset per half-wave

    for (int kc = 0; kc < F_IN; kc += 32) {
        __syncthreads();                          // lW reuse + A-panel ready
        // stage W chunk transposed: lW[k - kc][n] = W[n][k] (coalesced reads)
        {
            const int k  = kc + (threadIdx.x & 31);
            const int n0 = threadIdx.x >> 5;
            #pragma unroll
            for (int i = 0; i < 16; ++i) {
                const int n = n0 + 8 * i;
                lW[(k - kc) * SW + n] = W[n * F_IN + k];
            }
        }
        __syncthreads();

        #pragma unroll
        for (int kk = 0; kk < 32; kk += 4) {
            const int k0 = kk + kHalf;            // local K within chunk
            const int m  = lane & 15;
            v2f a, b;
            a[0] = lA[m * SA + kc + k0 + 0];
            a[1] = lA[m * SA + kc + k0 + 1];
            b[0] = lW[(k0 + 0) * SW + nCol];
            b[1] = lW[(k0 + 1) * SW + nCol];
            // D = A(16x4) x B(4x16) + C, fp32 throughout
            c = __builtin_amdgcn_wmma_f32_16x16x4_f32(
                    false, a, false, b, (short)0, c, false, false);
        }
    }

    // ---- bias -------------------------------------------------------------
    const float bv = bias[nCol];
    #pragma unroll
    for (int r = 0; r < 8; ++r) c[r] += bv;

    // ---- store D tile (layout: VGPR r -> M = r + 8*(lane/16), N = lane%16) -
    {
        const int mOff = (lane >> 4) * 8;
        #pragma unroll
        for (int r = 0; r < 8; ++r)
            Cout[(size_t)(rowBase + mOff + r) * F_OUT + nCol] = c[r];
    }

    // ---- BN statistics (layer-1 pass only) --------------------------------
    if (!FUSE) {
        float s = 0.0f, q = 0.0f;
        #pragma unroll
        for (int r = 0; r < 8; ++r) { s += c[r]; q += c[r] * c[r]; }
        s += __shfl_xor(s, 16, 32);   // combine the two half-waves (same N)
        q += __shfl_xor(q, 16, 32);
        if (lane < 16) {
            atomicAdd(&sumOut[nCol], s);
            atomicAdd(&sumsqOut[nCol], q);
        }
    }
}

// ---------------------------------------------------------------------------
// Kernel 4: fold BN stats into per-channel scale/shift.
//   scale = gamma * rsqrt(var+eps);  shift = beta - mu*scale
// ---------------------------------------------------------------------------
__global__ __launch_bounds__(128)
void gin_bnfold_kernel(const float* __restrict__ stats,  // sum[128], sumsq[128]
                       const float* __restrict__ gamma,
                       const float* __restrict__ beta,
                       float* __restrict__ scsh)         // scale[128], shift[128]
{
    const int ch = threadIdx.x;
    const float invN = 1.0f / (float)N_NODES;
    const float mu   = stats[ch] * invN;
    const float var  = stats[F_OUT + ch] * invN - mu * mu;
    const float sc   = __frsqrt_rn(var + BN_EPS) * gamma[ch];
    scsh[ch]         = sc;
    scsh[F_OUT + ch] = beta[ch] - mu * sc;
}

// ---------------------------------------------------------------------------
extern "C" void kernel_launch(void* const* d_in, const int* in_sizes, int n_in,
                              void* d_out, int out_size, void* d_ws, size_t ws_size,
                              hipStream_t stream)
{
    const float*     x     = (const float*)d_in[0];
    const long long* eidx  = (const long long*)d_in[1];   // int64 per reference
    const float*     eps   = (const float*)d_in[2];
    const float*     W1    = (const float*)d_in[3];
    const float*     b1    = (const float*)d_in[4];
    const float*     gamma = (const float*)d_in[5];
    const float*     beta  = (const float*)d_in[6];
    const float*     W2    = (const float*)d_in[7];
    const float*     b2    = (const float*)d_in[8];
    float*           out   = (float*)d_out;

    const long long* row = eidx;
    const long long* col = eidx + N_EDGES;

    // workspace: agg [N*128] | stats sum+sumsq [256] | scale+shift [256]
    float* agg   = (float*)d_ws;
    float* stats = agg + (size_t)N_NODES * F_IN;
    float* scsh  = stats + 2 * F_OUT;

    // 1) agg = (1+eps)*x ; zero BN stat accumulators
    {
        const int total4 = N_NODES * F_IN / 4;            // 3.2M float4
        gin_init_kernel<<<(total4 + 255) / 256, 256, 0, stream>>>(x, eps, agg, stats);
    }

    // 2) agg[row] += x[col]   (one wave per edge)
    {
        const long long threads = (long long)N_EDGES * 32;
        gin_scatter_kernel<<<(int)((threads + 255) / 256), 256, 0, stream>>>(x, row, col, agg);
    }

    // 3) h1 = agg @ W1^T + b1 -> d_out (scratch); fused BN stat accumulation
    gin_gemm_kernel<0><<<N_NODES / 16, 256, 0, stream>>>(
        agg, W1, b1, nullptr, nullptr, out, stats, stats + F_OUT);

    // 4) fold BN stats
    gin_bnfold_kernel<<<1, 128, 0, stream>>>(stats, gamma, beta, scsh);

    // 5) out = relu(bn(h1)) @ W2^T + b2, in place over d_out (block-local RAW only)
    gin_gemm_kernel<1><<<N_NODES / 16, 256, 0, stream>>>(
        out, W2, b2, scsh, scsh + F_OUT, out, nullptr, nullptr);
}